// MNN_70042326663664
// MI455X (gfx1250) — compile-verified
//
#include <hip/hip_runtime.h>

// ---------------------------------------------------------------------------
// Problem constants (from reference): x[64,3,64,64], K[16,3,7,7], fh=fw=58.
// Outputs: 3 tensors [64,48,58,58] (concat) = 31,002,624 f32 = 124 MB.
// Phase 1 is a 1.2G-term relu reduction to 32 scalars; phase 2 is a pure
// HBM-bandwidth broadcast fill (~5.3 us at 23.3 TB/s).
// ---------------------------------------------------------------------------

typedef __attribute__((ext_vector_type(2))) float v2f;
typedef __attribute__((ext_vector_type(8))) float v8f;

#define NPIX      786432      // 64*3*64*64
#define NCHUNK    24576       // NPIX / 32 (one wave-chunk = 32 consecutive pixels)
#define KSZ       2352        // 16*3*49 weights per tensor
#define TOTAL4    7750656     // total output float4s (3 * 64*48*58*58 / 4)
#define PT4       2583552     // float4s per output tensor

__global__ void zero32_kernel(float* s) {
    if (threadIdx.x < 32) s[threadIdx.x] = 0.0f;
}

// ---------------------------------------------------------------------------
// Phase 1: per-pixel reduction.
//   Weights staged in LDS as [c][dy*7+dx][o] so a wave (uniform c, uniform h,
//   w = w0+lane) does wave-uniform broadcast ds_load_b128 reads.
//   Boundary masking is branch-free: invalid (w-dx) taps use +/-3e38 sentinels
//   so relu() contributes exactly 0.
//   Block reduction of the 256x32 partial matrix uses exact-fp32 WMMA:
//   D = ones(16x4) x B(4x16) + C  => column sums, chained over 64 tiles/group.
// ---------------------------------------------------------------------------
__global__ __launch_bounds__(256) void mnn_phase1_kernel(
    const float* __restrict__ x,
    const float* __restrict__ Khit,
    const float* __restrict__ Kmiss,
    float* __restrict__ sums)     // sums[0..15]=S_hit, sums[16..31]=S_miss
{
    __shared__ float sKh[3 * 49 * 16];      // [c][tap][o]
    __shared__ float sKm[3 * 49 * 16];
    __shared__ float part[256 * 33];        // padded stride 33: conflict-free

    const int tid = threadIdx.x;

    // Stage weights into LDS, transposed to [c][tap][o].
    for (int i = tid; i < KSZ; i += 256) {
        int o   = i / 147;          // K layout is [o][c][dy][dx]
        int cp  = i - o * 147;      // c*49 + tap
        sKh[cp * 16 + o] = Khit[i];
        sKm[cp * 16 + o] = Kmiss[i];
    }
    __syncthreads();

    float accH[16], accM[16];
#pragma unroll
    for (int o = 0; o < 16; ++o) { accH[o] = 0.0f; accM[o] = 0.0f; }

    const int lane   = tid & 31;
    const int waveId = tid >> 5;
    const int gwave  = blockIdx.x * 8 + waveId;
    const int nwaves = gridDim.x * 8;

    for (int chunk = gwave; chunk < NCHUNK; chunk += nwaves) {
        const int p  = chunk * 32 + lane;
        const float xv = x[p];
        __builtin_prefetch(&x[p + nwaves * 32], 0, 0);   // global_prefetch_b8
        const int c = (p >> 12) % 3;    // wave-uniform
        const int h = (p >> 6) & 63;    // wave-uniform (32 | chunk base, W=64)
        const int w = p & 63;

        const float* khb = &sKh[c * 49 * 16];
        const float* kmb = &sKm[c * 49 * 16];

#pragma unroll
        for (int dy = 0; dy < 7; ++dy) {
            if ((unsigned)(h - dy) > 57u) continue;      // uniform skip
#pragma unroll
            for (int dx = 0; dx < 7; ++dx) {
                const bool valid = (unsigned)(w - dx) <= 57u;
                const float xh = valid ? xv :  3.0e38f;  // relu(K - 3e38) == 0
                const float xm = valid ? xv : -3.0e38f;  // relu(-3e38 - K) == 0
                const int tap = (dy * 7 + dx) * 16;
                const float4* kh4 = reinterpret_cast<const float4*>(khb + tap);
                const float4* km4 = reinterpret_cast<const float4*>(kmb + tap);
#pragma unroll
                for (int q = 0; q < 4; ++q) {            // ds_load_b128 broadcasts
                    const float4 kh = kh4[q];
                    const float4 km = km4[q];
                    accH[4*q+0] += fmaxf(kh.x - xh, 0.0f);
                    accH[4*q+1] += fmaxf(kh.y - xh, 0.0f);
                    accH[4*q+2] += fmaxf(kh.z - xh, 0.0f);
                    accH[4*q+3] += fmaxf(kh.w - xh, 0.0f);
                    accM[4*q+0] += fmaxf(xm - km.x, 0.0f);
                    accM[4*q+1] += fmaxf(xm - km.y, 0.0f);
                    accM[4*q+2] += fmaxf(xm - km.z, 0.0f);
                    accM[4*q+3] += fmaxf(xm - km.w, 0.0f);
                }
            }
        }
    }

#pragma unroll
    for (int o = 0; o < 16; ++o) {
        part[tid * 33 + o]      = accH[o];
        part[tid * 33 + 16 + o] = accM[o];
    }
    __syncthreads();

    // Wave 0: WMMA column-sum reduction of part[256][32] -> 32 scalars.
    // A = ones(16x4) exact in fp32; C accumulates across 64 chained WMMAs.
    if (waveId == 0) {
        const v2f A = {1.0f, 1.0f};
#pragma unroll
        for (int g = 0; g < 2; ++g) {               // g=0: hit, g=1: miss
            v8f acc = {0.f, 0.f, 0.f, 0.f, 0.f, 0.f, 0.f, 0.f};
            const int col  = g * 16 + (lane & 15);  // B col N = lane&15
            const int krow = (lane >> 4) * 2;       // B rows: K=0,1 | K=2,3
            for (int rb = 0; rb < 256; rb += 4) {
                v2f Bv;
                Bv.x = part[(rb + krow + 0) * 33 + col];
                Bv.y = part[(rb + krow + 1) * 33 + col];
                acc = __builtin_amdgcn_wmma_f32_16x16x4_f32(
                    /*neg_a=*/false, A, /*neg_b=*/false, Bv,
                    /*c_mod=*/(short)0, acc, /*reuse_a=*/false, /*reuse_b=*/false);
            }
            // Every D row equals the column sums; row 0 lives in vgpr 0.
            if (lane < 16) atomicAdd(&sums[g * 16 + lane], acc[0]);
        }
    }
}

// ---------------------------------------------------------------------------
// Phase 2: broadcast fill of the three output tensors with 128-bit stores.
// Channel j of a tensor maps to output channel o = j/3; each (b,j) plane is
// 3364 floats = 841 float4s, so float4s never straddle a channel boundary.
// ---------------------------------------------------------------------------
__global__ __launch_bounds__(256) void mnn_phase2_kernel(
    const float* __restrict__ sums, float4* __restrict__ out)
{
    __shared__ float vals[3][16];
    const int tid = threadIdx.x;
    if (tid < 16) {
        const float Sh = sums[tid];        // sum relu(K_hit - x)  (positive)
        const float Sm = sums[16 + tid];   // sum relu(x - K_miss)
        vals[0][tid] = -Sh - Sm;           // F_map   = s_hit - s_miss
        vals[1][tid] = -Sh;                // F_hit_list
        vals[2][tid] =  Sm;                // F_miss_list
    }
    __syncthreads();

    for (int i = blockIdx.x * 256 + tid; i < TOTAL4; i += gridDim.x * 256) {
        const int t     = i / PT4;
        const int r     = i - t * PT4;
        const int chunk = r / 841;         // b*48 + j
        const int j     = chunk % 48;
        const float v   = vals[t][j / 3];
        out[i] = make_float4(v, v, v, v);
    }
}

extern "C" void kernel_launch(void* const* d_in, const int* in_sizes, int n_in,
                              void* d_out, int out_size, void* d_ws, size_t ws_size,
                              hipStream_t stream) {
    const float* x     = (const float*)d_in[0];
    const float* Khit  = (const float*)d_in[1];
    const float* Kmiss = (const float*)d_in[2];
    float* sums = (float*)d_ws;            // 32 f32 scratch accumulators

    zero32_kernel<<<1, 32, 0, stream>>>(sums);
    mnn_phase1_kernel<<<2048, 256, 0, stream>>>(x, Khit, Kmiss, sums);
    mnn_phase2_kernel<<<8192, 256, 0, stream>>>(sums, (float4*)d_out);
}